// VectorQuantiser_50199577756002
// MI455X (gfx1250) — compile-verified
//
#include <hip/hip_runtime.h>
#include <stdint.h>

#define B_  16
#define T_  4096
#define D_  128
#define K_  1024
#define BT_ (B_ * T_)   // 65536

#define ZQ_OFF_C   ((size_t)0)
#define SIM_OFF_C  ((size_t)BT_ * D_)                 // 8,388,608
#define IDS_OFF_C  (SIM_OFF_C + (size_t)BT_ * K_)     // 75,497,472
#define LOSS_OFF_C (IDS_OFF_C + (size_t)BT_)          // 75,563,008

typedef float v2f __attribute__((ext_vector_type(2)));
typedef float v8f __attribute__((ext_vector_type(8)));

static __device__ __forceinline__ float wave_reduce_add(float x) {
#pragma unroll
  for (int off = 16; off > 0; off >>= 1) x += __shfl_xor(x, off, 32);
  return x;
}

// ---------------------------------------------------------------------------
// CDNA5 async copy: global -> LDS, 16B per lane, tracked by ASYNCcnt.
// GV mode: %0 = per-lane LDS byte offset VGPR, %1 = 64-bit global address.
// ---------------------------------------------------------------------------
static __device__ __forceinline__ void async_g2lds_b128(uint32_t lds_off,
                                                        const void* gsrc) {
  asm volatile("global_load_async_to_lds_b128 %0, %1, off"
               :
               : "v"(lds_off), "v"((uint64_t)(uintptr_t)gsrc)
               : "memory");
}

static __device__ __forceinline__ void wait_async0() {
  asm volatile("s_wait_asynccnt 0x0" ::: "memory");
}

// ---------------------------------------------------------------------------
// Prep: per-code squared norms + rsqrt, and zero the loss accumulator.
// One wave per code (K=1024 -> 128 blocks x 8 waves), 4 floats per lane.
// ---------------------------------------------------------------------------
__global__ __launch_bounds__(256) void vq_prep(const float* __restrict__ E,
                                               float* __restrict__ nsq_e,
                                               float* __restrict__ rse,
                                               float* __restrict__ loss_slot) {
  const int lane = threadIdx.x & 31;
  const int wave = threadIdx.x >> 5;
  const int k = blockIdx.x * 8 + wave;
  const float4 e4 = *(const float4*)(E + (size_t)k * D_ + lane * 4);
  float s = e4.x * e4.x + e4.y * e4.y + e4.z * e4.z + e4.w * e4.w;
  s = wave_reduce_add(s);
  if (lane == 0) {
    nsq_e[k] = s;
    rse[k]   = rsqrtf(s);
  }
  if (blockIdx.x == 0 && threadIdx.x == 0) *loss_slot = 0.0f;
}

// ---------------------------------------------------------------------------
// Main: 8 waves/block, 16 rows/wave (128 rows/block, grid = 512 blocks).
// A (16x128 of z_e) lives in registers as v_wmma f32 16x16x4 fragments.
// E staged via double-buffered GLOBAL_LOAD_ASYNC_TO_LDS_B128 (32-code chunks,
// rows padded to 132 floats -> lanes hit banks {4n,4n+1}/{4n+2,4n+3}).
// ---------------------------------------------------------------------------
#define CHUNK_CODES 32
#define NCHUNKS     (K_ / CHUNK_CODES)   // 32
#define LDS_STRIDE  132

static __device__ __forceinline__ void stage_chunk_async(
    const float* __restrict__ E, int c, float* lds_buf, int tid) {
  const float* src = E + (size_t)c * CHUNK_CODES * D_;
  // 32 codes x 128 floats = 1024 float4 -> 4 per thread, coalesced.
#pragma unroll
  for (int j = 0; j < 4; ++j) {
    const int idx4 = tid + j * 256;
    const int p    = idx4 * 4;     // float index inside chunk
    const int code = p >> 7;       // / 128
    const int d    = p & 127;
    const uint32_t lds_off =
        (uint32_t)(uintptr_t)(lds_buf + code * LDS_STRIDE + d);
    async_g2lds_b128(lds_off, src + p);
  }
}

__global__ __launch_bounds__(256) void vq_main(const float* __restrict__ Z,
                                               const float* __restrict__ E,
                                               const float* __restrict__ nsq_e,
                                               const float* __restrict__ rse,
                                               float* __restrict__ out) {
  __shared__ float lds_e[2][CHUNK_CODES * LDS_STRIDE];  // 2 x 16.5 KB
  __shared__ int   s_ids[8][16];

  const int lane = threadIdx.x & 31;
  const int wave = threadIdx.x >> 5;
  const int hi   = lane >> 4;      // half-wave select (0/1)
  const int lmod = lane & 15;
  const int ksel = hi * 2;         // A/B fragment k sub-select per ISA layout
  const int row0 = blockIdx.x * 128 + wave * 16;

  // ---- Load A fragments (whole D=128 -> 32 k-steps of 4) + row norm -------
  v2f a[32];
  const float* zrow = Z + (size_t)(row0 + lmod) * D_;
  float nz_part = 0.0f;
#pragma unroll
  for (int s = 0; s < 32; ++s) {
    float2 t = *(const float2*)(zrow + 4 * s + ksel);
    a[s].x = t.x;
    a[s].y = t.y;
    nz_part += t.x * t.x + t.y * t.y;
  }
  const float nz  = nz_part + __shfl_xor(nz_part, 16, 32);  // pair (l, l^16)
  const float rsn = rsqrtf(nz);
  // rsn for the 8 rows this lane sees in the accumulator (rows v + 8*hi)
  float rsn8[8];
#pragma unroll
  for (int v = 0; v < 8; ++v) rsn8[v] = __shfl(rsn, v + 8 * hi, 32);

  float best[8];
  int   bidx[8];
#pragma unroll
  for (int v = 0; v < 8; ++v) {
    best[v] = 3.4e38f;
    bidx[v] = 0;
  }

  // ---- K loop: 32 double-buffered chunks of 32 codes ----------------------
  stage_chunk_async(E, 0, &lds_e[0][0], threadIdx.x);

  for (int c = 0; c < NCHUNKS; ++c) {
    wait_async0();     // own async copies into lds_e[c&1] done
    __syncthreads();   // ...and everyone else's
    if (c + 1 < NCHUNKS)
      stage_chunk_async(E, c + 1, &lds_e[(c + 1) & 1][0], threadIdx.x);

    float* buf = &lds_e[c & 1][0];
#pragma unroll
    for (int ct = 0; ct < 2; ++ct) {
      const int col0 = c * CHUNK_CODES + ct * 16;
      const float ne_s  = nsq_e[col0 + lmod];
      const float rse_s = rse[col0 + lmod];
      const float* brow = &buf[(ct * 16 + lmod) * LDS_STRIDE + ksel];

      v8f acc = {};
#pragma unroll
      for (int s = 0; s < 32; ++s) {
        float2 t = *(const float2*)(brow + 4 * s);
        v2f b;
        b.x = t.x;
        b.y = t.y;
        // D = A(16x4) x B(4x16) + C  -- f32 WMMA, exact-precision path
        acc = __builtin_amdgcn_wmma_f32_16x16x4_f32(
            /*neg_a=*/false, a[s], /*neg_b=*/false, b,
            /*c_mod=*/(short)0, acc, /*reuse_a=*/false, /*reuse_b=*/false);
      }

      // Post-process tile: similarity (streaming NT store) + running argmin
#pragma unroll
      for (int v = 0; v < 8; ++v) {
        const int row = row0 + v + 8 * hi;
        __builtin_nontemporal_store(
            acc[v] * rsn8[v] * rse_s,
            out + SIM_OFF_C + (size_t)row * K_ + col0 + lmod);
        const float score = ne_s - 2.0f * acc[v];  // nz is row-constant
        if (score < best[v]) {
          best[v] = score;
          bidx[v] = col0 + lmod;
        }
      }
    }
  }

  // ---- Argmin reduction across the 16 lanes of each half-wave -------------
#pragma unroll
  for (int v = 0; v < 8; ++v) {
#pragma unroll
    for (int off = 8; off >= 1; off >>= 1) {
      const float ov = __shfl_xor(best[v], off, 32);
      const int   oi = __shfl_xor(bidx[v], off, 32);
      if (ov < best[v] || (ov == best[v] && oi < bidx[v])) {
        best[v] = ov;
        bidx[v] = oi;
      }
    }
    if (lmod == 0) {
      const int r = v + 8 * hi;
      out[IDS_OFF_C + row0 + r] = (float)bidx[v];
      s_ids[wave][r] = bidx[v];
    }
  }
  __syncthreads();

  // ---- Gather z_q = E[ids] and per-row L2 loss ----------------------------
  float wave_loss = 0.0f;
#pragma unroll 1
  for (int r = 0; r < 16; ++r) {
    const int id  = s_ids[wave][r];
    const int row = row0 + r;
    const float4 e4 = *(const float4*)(E + (size_t)id * D_ + lane * 4);
    *(float4*)(out + ZQ_OFF_C + (size_t)row * D_ + lane * 4) = e4;
    const float4 z4 = *(const float4*)(Z + (size_t)row * D_ + lane * 4);
    const float dx = z4.x - e4.x, dy = z4.y - e4.y;
    const float dz = z4.z - e4.z, dw = z4.w - e4.w;
    float ds = dx * dx + dy * dy + dz * dz + dw * dw;
    ds = wave_reduce_add(ds);
    if (lane == 0) wave_loss += sqrtf(ds);
  }
  // loss = loss_codebook + BETA*loss_commit = 1.25 * mean(||z_e - z_q||)
  if (lane == 0) atomicAdd(out + LOSS_OFF_C, wave_loss * (1.25f / (float)BT_));
}

// ---------------------------------------------------------------------------
extern "C" void kernel_launch(void* const* d_in, const int* in_sizes, int n_in,
                              void* d_out, int out_size, void* d_ws,
                              size_t ws_size, hipStream_t stream) {
  const float* z_e = (const float*)d_in[0];  // [B,T,D]
  const float* E   = (const float*)d_in[1];  // [K,D]
  float* out = (float*)d_out;
  float* nsq_e = (float*)d_ws;   // K floats
  float* rse   = nsq_e + K_;     // K floats

  vq_prep<<<K_ / 8, 256, 0, stream>>>(E, nsq_e, rse, out + LOSS_OFF_C);
  vq_main<<<BT_ / 128, 256, 0, stream>>>(z_e, E, nsq_e, rse, out);
}